// NonParamBesselRBF_18820546691339
// MI455X (gfx1250) — compile-verified
//
#include <hip/hip_runtime.h>
#include <math.h>

typedef __attribute__((ext_vector_type(2))) float v2f;
typedef __attribute__((ext_vector_type(8))) float v8f;

#define DIMK 64      // basis dimension
#define OUTD 64      // output channels

// Each block: 256 threads = 8 waves; each lane owns one (i,j) pair -> 256 pairs
// per block, all sharing the same i (N is a multiple of 256).
__global__ __launch_bounds__(256) void NonParamBesselRBF_kernel(
    const float* __restrict__ atoms,
    const float* __restrict__ w1,   // (64,4) row-major
    const float* __restrict__ w2,   // (4,64) row-major
    float* __restrict__ out,        // (N,N,64)
    int N, int blocksPerRow)
{
    __shared__ float4 w1row[DIMK];          // w1 row d = float4(w1[d][0..3])

    const int tid = threadIdx.x;
    if (tid < DIMK) {
        w1row[tid] = ((const float4*)w1)[tid];
    }
    __syncthreads();

    const int i      = blockIdx.x / blocksPerRow;
    const int jblock = (blockIdx.x % blocksPerRow) * 256;
    const int j      = jblock + tid;

    const int lane = tid & 31;
    const int half = lane >> 4;             // 0: lanes 0-15, 1: lanes 16-31
    const int n    = lane & 15;

    // ---- pairwise distance (atoms[i] is block-uniform -> scalarized) ----
    const float xi = atoms[i*3+0], yi = atoms[i*3+1], zi = atoms[i*3+2];
    const float xj = atoms[j*3+0], yj = atoms[j*3+1], zj = atoms[j*3+2];
    const float dx = xi - xj, dy = yi - yj, dz = zi - zj;
    const float x  = sqrtf(dx*dx + dy*dy + dz*dz) + 1e-8f;   // x_ext

    // basis_d = sqrt(2/5) * sin((d+1)*t) / x,  t = pi*x/5
    const float t   = x * 0.6283185307179586f;               // pi/5
    float s0, c0;
    sincosf(t, &s0, &c0);                                    // one accurate sincos/pair
    const float inv  = 0.6324555320336759f / x;              // sqrt(2/5)/x
    const float twoc = 2.0f * c0;

    // ---- rank-4 contraction: p_c = sum_d basis_d * w1[d][c], via recurrence ----
    float sp = 0.0f, sc = s0;                // sin(0*t), sin(1*t)
    float p0 = 0.f, p1 = 0.f, p2 = 0.f, p3 = 0.f;
    #pragma unroll 4
    for (int d = 0; d < DIMK; ++d) {
        const float4 w = w1row[d];           // ds_load_b128, broadcast
        const float  b = sc * inv;
        p0 = fmaf(b, w.x, p0);
        p1 = fmaf(b, w.y, p1);
        p2 = fmaf(b, w.z, p2);
        p3 = fmaf(b, w.w, p3);
        const float sn = fmaf(twoc, sc, -sp);  // sin((d+2)t)
        sp = sc; sc = sn;
    }

    // ---- assemble WMMA A tiles (16x4 f32): lanes 0-15 hold K=0,1; 16-31 hold K=2,3
    const float p0x = __shfl_xor(p0, 16, 32);
    const float p1x = __shfl_xor(p1, 16, 32);
    const float p2x = __shfl_xor(p2, 16, 32);
    const float p3x = __shfl_xor(p3, 16, 32);

    v2f a0, a1;
    a0.x = half ? p2x : p0;  a0.y = half ? p3x : p1;   // pairs 0-15 of the wave
    a1.x = half ? p2  : p0x; a1.y = half ? p3  : p1x;  // pairs 16-31 of the wave

    const int k0 = half * 2;                 // B rows held by this lane half
    const v8f zero = {};
    const int jw = jblock + (tid & ~31);     // wave's first pair (j index)

    #pragma unroll
    for (int q = 0; q < 4; ++q) {            // 4 chunks of 16 output channels
        v2f bm;                              // B (4x16): w2[k][q*16+n]
        bm.x = w2[(k0    ) * OUTD + q*16 + n];
        bm.y = w2[(k0 + 1) * OUTD + q*16 + n];

        // D = A x B (+0): 16 pairs x 16 out-channels, f32 accumulate
        v8f d0 = __builtin_amdgcn_wmma_f32_16x16x4_f32(
            false, a0, false, bm, (short)0, zero, false, false);
        v8f d1 = __builtin_amdgcn_wmma_f32_16x16x4_f32(
            false, a1, false, bm, (short)0, zero, false, false);

        // D layout: VGPR r holds rows r (lanes 0-15) and r+8 (lanes 16-31)
        const unsigned base0 =
            ((unsigned)i * (unsigned)N + (unsigned)(jw + 8*half)) * OUTD
            + (unsigned)(q*16 + n);
        #pragma unroll
        for (int r = 0; r < 8; ++r) {
            __builtin_nontemporal_store(d0[r], out + base0 + (unsigned)r        * OUTD);
            __builtin_nontemporal_store(d1[r], out + base0 + (unsigned)(16 + r) * OUTD);
        }
    }
}

extern "C" void kernel_launch(void* const* d_in, const int* in_sizes, int n_in,
                              void* d_out, int out_size, void* d_ws, size_t ws_size,
                              hipStream_t stream) {
    const float* atoms = (const float*)d_in[0];
    const float* w1    = (const float*)d_in[1];
    const float* w2    = (const float*)d_in[2];
    float*       out   = (float*)d_out;

    const int N   = in_sizes[0] / 3;         // 2048
    const int bpr = N / 256;                 // blocks per atom-row (8)
    const int blocks = N * bpr;              // 16384 blocks of 256 threads

    NonParamBesselRBF_kernel<<<blocks, 256, 0, stream>>>(atoms, w1, w2, out, N, bpr);
}